// Loss_12232066859491
// MI455X (gfx1250) — compile-verified
//
#include <hip/hip_runtime.h>

typedef __attribute__((ext_vector_type(2))) float v2f;
typedef __attribute__((ext_vector_type(8))) float v8f;

#define NIMG 117            // 9 * 13 images
#define SSIM_C1 1e-4f       // (0.01*1)^2
#define SSIM_C2 9e-4f       // (0.03*1)^2

__constant__ float c_lo[9] = {170.0f, 85000.0f, -110.0f, -80.0f, 170.0f, 0.0f, -110.0f, -100.0f, -1000.0f};
__constant__ float c_hi[9] = {350.0f, 110000.0f, 110.0f, 80.0f, 350.0f, 0.04f, 110.0f, 100.0f, 60000.0f};

__device__ __forceinline__ float norm01(float v, float lo, float isp) {
    return fminf(fmaxf((v - lo) * isp, 0.f), 1.f);
}

// ---------------------------------------------------------------------------
// Fused SSIM kernel for one pyramid scale.
// Each 256-thread block (8 waves) computes a 32x64 tile of the (H-10)x(W-10)
// valid conv region for one image. Each wave owns a 16x16 output tile and
// evaluates the separable 11-tap Gaussian for 5 quantities (X, Y, X^2, Y^2,
// X*Y) entirely with V_WMMA_F32_16X16X4_F32 banded matmuls.
// ---------------------------------------------------------------------------
#define LDS_H 44
#define LDS_W 80

__global__ __launch_bounds__(256) void ssim_kernel(
    const float* __restrict__ X, const float* __restrict__ Y,
    int H, int W, int validH, int validW, int tilesX, int normalize,
    float* __restrict__ accSS, float* __restrict__ accCS)
{
    __shared__ float sX[LDS_H][LDS_W];
    __shared__ float sY[LDS_H][LDS_W];
    __shared__ float sV[8][16][32];   // per-wave vertical-filtered scratch
    __shared__ float sW[48];          // band coeffs: sW[16+k]=w[k] for k in [0,10], else 0
    __shared__ float red0[256];
    __shared__ float red1[256];

    const int tid    = threadIdx.x;
    const int img    = blockIdx.y;
    const int tileId = blockIdx.x;
    const int tX = tileId % tilesX;
    const int tY = tileId / tilesX;
    const int wgR0 = tY * 32;
    const int wgC0 = tX * 64;
    const size_t imgBase = (size_t)img * H * W;

    // Gaussian window (sigma=1.5, 11 taps), zero-padded lookup table.
    if (tid < 48) {
        float v = 0.f;
        int k = tid - 16;
        if (k >= 0 && k <= 10) {
            float s = 0.f;
            for (int i = 0; i < 11; ++i) { float d = (float)i - 5.f; s += expf(-d * d / 4.5f); }
            float d = (float)k - 5.f;
            v = expf(-d * d / 4.5f) / s;
        }
        sW[tid] = v;
    }

    float lo = 0.f, isp = 1.f;
    if (normalize) { int n = img / 13; lo = c_lo[n]; isp = 1.f / (c_hi[n] - c_lo[n]); }

    // Cooperative stage of the (padded) input tile; out-of-image -> 0.
    for (int idx = tid; idx < LDS_H * LDS_W; idx += 256) {
        int rr = idx / LDS_W, cc = idx % LDS_W;
        int gr = wgR0 + rr, gc = wgC0 + cc;
        float xv = 0.f, yv = 0.f;
        if (gr < H && gc < W) {
            size_t o = imgBase + (size_t)gr * W + gc;
            xv = X[o]; yv = Y[o];
            if (normalize) { xv = norm01(xv, lo, isp); yv = norm01(yv, lo, isp); }
        }
        sX[rr][cc] = xv;
        sY[rr][cc] = yv;
    }
    __syncthreads();

    const int lane = tid & 31;
    const int wid  = tid >> 5;
    const int r0   = (wid >> 2) * 16;   // wave tile row offset in workgroup tile
    const int c0   = (wid & 3) * 16;    // wave tile col offset
    const int M    = lane & 15;
    const int koff = (lane & 16) ? 2 : 0;
    const int mhi  = (lane & 16) ? 8 : 0;

    v8f F[5];
    #pragma unroll
    for (int q = 0; q < 5; ++q) {
        // Vertical pass: Vout(16x26) = Wv(16x28) x Xq(28x16) for two N-groups.
        #pragma unroll
        for (int g = 0; g < 2; ++g) {
            v8f acc = {0.f, 0.f, 0.f, 0.f, 0.f, 0.f, 0.f, 0.f};
            const int ccol = c0 + 16 * g + M;
            #pragma unroll
            for (int kk = 0; kk < 7; ++kk) {
                const int K0 = 4 * kk + koff;
                v2f a, b;
                a.x = sW[16 + K0 - M];       // Wv[m][k] = w[k-m], zero outside band
                a.y = sW[17 + K0 - M];
                const int rrow = r0 + K0;
                float x0 = sX[rrow][ccol],     y0 = sY[rrow][ccol];
                float x1 = sX[rrow + 1][ccol], y1 = sY[rrow + 1][ccol];
                b.x = (q == 0) ? x0 : (q == 1) ? y0 : (q == 2) ? x0 * x0 : (q == 3) ? y0 * y0 : x0 * y0;
                b.y = (q == 0) ? x1 : (q == 1) ? y1 : (q == 2) ? x1 * x1 : (q == 3) ? y1 * y1 : x1 * y1;
                acc = __builtin_amdgcn_wmma_f32_16x16x4_f32(false, a, false, b, (short)0, acc, false, false);
            }
            #pragma unroll
            for (int r = 0; r < 8; ++r)
                sV[wid][r + mhi][M + 16 * g] = acc[r];
        }
        // Horizontal pass: Out(16x16) = Vout(16x28) x Wh(28x16).
        v8f h = {0.f, 0.f, 0.f, 0.f, 0.f, 0.f, 0.f, 0.f};
        #pragma unroll
        for (int kk = 0; kk < 7; ++kk) {
            const int K0 = 4 * kk + koff;
            v2f a, b;
            a.x = sV[wid][M][K0];
            a.y = sV[wid][M][K0 + 1];
            b.x = sW[16 + K0 - M];           // Wh[p][n] = w[p-n]
            b.y = sW[17 + K0 - M];
            h = __builtin_amdgcn_wmma_f32_16x16x4_f32(false, a, false, b, (short)0, h, false, false);
        }
        F[q] = h;
    }

    // SSIM / CS maps + masked accumulation.
    float sumSS = 0.f, sumCS = 0.f;
    #pragma unroll
    for (int r = 0; r < 8; ++r) {
        int orow = wgR0 + r0 + r + mhi;
        int ocol = wgC0 + c0 + M;
        float mu1 = F[0][r], mu2 = F[1][r];
        float s1  = F[2][r] - mu1 * mu1;
        float s2  = F[3][r] - mu2 * mu2;
        float s12 = F[4][r] - mu1 * mu2;
        float cs = (2.f * s12 + SSIM_C2) / (s1 + s2 + SSIM_C2);
        float ss = ((2.f * mu1 * mu2 + SSIM_C1) / (mu1 * mu1 + mu2 * mu2 + SSIM_C1)) * cs;
        bool valid = (orow < validH) && (ocol < validW);
        sumSS += valid ? ss : 0.f;
        sumCS += valid ? cs : 0.f;
    }
    red0[tid] = sumSS;
    red1[tid] = sumCS;
    __syncthreads();
    for (int s = 128; s > 0; s >>= 1) {
        if (tid < s) { red0[tid] += red0[tid + s]; red1[tid] += red1[tid + s]; }
        __syncthreads();
    }
    if (tid == 0) {
        atomicAdd(&accSS[img], red0[0]);
        atomicAdd(&accCS[img], red1[0]);
    }
}

// ---------------------------------------------------------------------------
// 2x2 average pool with jax.lax.reduce_window semantics (zero-pad odd H by 1
// on each side, sum * 0.25). Optionally normalizes raw inputs on the fly.
// ---------------------------------------------------------------------------
__global__ __launch_bounds__(256) void pool_kernel(
    const float* __restrict__ X, const float* __restrict__ Y,
    float* __restrict__ OX, float* __restrict__ OY,
    int Hi, int Wi, int Ho, int Wo, int normalize)
{
    long long total = (long long)NIMG * Ho * Wo;
    long long id = (long long)blockIdx.x * 256 + threadIdx.x;
    if (id >= total) return;
    int hw = Ho * Wo;
    int img = (int)(id / hw);
    int rem = (int)(id % hw);
    int i = rem / Wo, j = rem % Wo;
    size_t base = (size_t)img * Hi * Wi;
    int rA = 2 * i - 1, rB = 2 * i;      // ph = 1 (H odd at every pooled scale)
    int cA = 2 * j, cB = 2 * j + 1;      // pw = 0 (W even)
    float lo = 0.f, isp = 1.f;
    if (normalize) { int n = img / 13; lo = c_lo[n]; isp = 1.f / (c_hi[n] - c_lo[n]); }
    float sx = 0.f, sy = 0.f;
    if (rA >= 0) {
        float a = X[base + (size_t)rA * Wi + cA], b = X[base + (size_t)rA * Wi + cB];
        float c = Y[base + (size_t)rA * Wi + cA], d = Y[base + (size_t)rA * Wi + cB];
        if (normalize) { a = norm01(a, lo, isp); b = norm01(b, lo, isp); c = norm01(c, lo, isp); d = norm01(d, lo, isp); }
        sx += a + b; sy += c + d;
    }
    {
        float a = X[base + (size_t)rB * Wi + cA], b = X[base + (size_t)rB * Wi + cB];
        float c = Y[base + (size_t)rB * Wi + cA], d = Y[base + (size_t)rB * Wi + cB];
        if (normalize) { a = norm01(a, lo, isp); b = norm01(b, lo, isp); c = norm01(c, lo, isp); d = norm01(d, lo, isp); }
        sx += a + b; sy += c + d;
    }
    OX[id] = 0.25f * sx;
    OY[id] = 0.25f * sy;
}

// ---------------------------------------------------------------------------
// Fused pixel loss: sum(w*|d|) and sum((1-w)*d^2), float4-vectorized.
// ---------------------------------------------------------------------------
__global__ __launch_bounds__(256) void pixel_kernel(
    const float* __restrict__ X, const float* __restrict__ Y,
    float* __restrict__ accAbs, float* __restrict__ accSq)
{
    __shared__ float r0[256];
    __shared__ float r1[256];
    const long long n4 = 30368520LL;            // 121474080 / 4
    long long id = (long long)blockIdx.x * 256 + threadIdx.x;
    float sa = 0.f, sq = 0.f;
    if (id < n4) {
        int img = (int)(id / 259560LL);         // float4 groups per image
        int n = img / 13;
        float lo = c_lo[n], isp = 1.f / (c_hi[n] - c_lo[n]);
        float4 xv = ((const float4*)X)[id];
        float4 yv = ((const float4*)Y)[id];
        float xs[4] = {xv.x, xv.y, xv.z, xv.w};
        float ys[4] = {yv.x, yv.y, yv.z, yv.w};
        #pragma unroll
        for (int k = 0; k < 4; ++k) {
            float xr = norm01(xs[k], lo, isp);
            float yr = norm01(ys[k], lo, isp);
            float d = xr - yr;
            float w = expf(5.f * yr * yr * yr) + 1.f;
            sa += w * fabsf(d);
            sq += (1.f - w) * d * d;
        }
    }
    r0[threadIdx.x] = sa;
    r1[threadIdx.x] = sq;
    __syncthreads();
    for (int s = 128; s > 0; s >>= 1) {
        if (threadIdx.x < s) { r0[threadIdx.x] += r0[threadIdx.x + s]; r1[threadIdx.x] += r1[threadIdx.x + s]; }
        __syncthreads();
    }
    if (threadIdx.x == 0) { atomicAdd(accAbs, r0[0]); atomicAdd(accSq, r1[0]); }
}

// ---------------------------------------------------------------------------
// Final combine: per-image MS-SSIM product, mean over images, + pixel loss.
// ---------------------------------------------------------------------------
__global__ __launch_bounds__(128) void final_kernel(const float* __restrict__ acc, float* __restrict__ out)
{
    __shared__ float red[128];
    const float cnt[5] = {1016730.f, 249210.f, 59850.f, 13770.f, 2880.f}; // valid pixels per scale
    const float wt[5]  = {0.0448f, 0.2856f, 0.3001f, 0.2363f, 0.1333f};
    float s = 0.f;
    for (int img = threadIdx.x; img < NIMG; img += 128) {
        float ms = 1.f;
        #pragma unroll
        for (int sc = 0; sc < 5; ++sc) {
            float ssm = acc[sc * NIMG + img] / cnt[sc];
            float csm = acc[1024 + sc * NIMG + img] / cnt[sc];
            float v = (sc < 4) ? csm : ssm;     // cs at scales 0..3, ssim at scale 4
            v = fmaxf(v, 0.f);
            ms *= powf(v, wt[sc]);
        }
        s += ms;
    }
    red[threadIdx.x] = s;
    __syncthreads();
    for (int t = 64; t > 0; t >>= 1) {
        if (threadIdx.x < t) red[threadIdx.x] += red[threadIdx.x + t];
        __syncthreads();
    }
    if (threadIdx.x == 0) {
        float ms_mean = red[0] / (float)NIMG;
        const float NT = 121474080.f;
        float pixel = 0.5f * acc[2048] / NT + 0.5f * acc[2049] / NT;
        out[0] = (1.f - ms_mean) + pixel;
    }
}

extern "C" void kernel_launch(void* const* d_in, const int* in_sizes, int n_in,
                              void* d_out, int out_size, void* d_ws, size_t ws_size,
                              hipStream_t stream)
{
    const float* x = (const float*)d_in[0];
    const float* y = (const float*)d_in[1];
    float* ws  = (float*)d_ws;
    float* out = (float*)d_out;

    // Workspace layout (floats):
    //   [0, 4096)            accumulators (ssim sums, cs sums, pixel sums)
    //   then pyramid buffers for scales 1..4 (x and y each)
    const long long L1 = 30410640LL;  // 117*361*720
    const long long L2 = 7623720LL;   // 117*181*360
    const long long L3 = 1916460LL;   // 117*91*180
    const long long L4 = 484380LL;    // 117*46*90
    float* acc = ws;
    float* s1x = ws + 4096;    float* s1y = s1x + L1;
    float* s2x = s1y + L1;     float* s2y = s2x + L2;
    float* s3x = s2y + L2;     float* s3y = s3x + L3;
    float* s4x = s3y + L3;     float* s4y = s4x + L4;

    hipMemsetAsync(acc, 0, 4096 * sizeof(float), stream);

    // Pixel loss (reads raw inputs once).
    {
        long long n4 = 30368520LL;
        pixel_kernel<<<(int)((n4 + 255) / 256), 256, 0, stream>>>(x, y, acc + 2048, acc + 2049);
    }

    // Build the normalized pyramid.
    {
        long long t;
        t = (long long)NIMG * 361 * 720;
        pool_kernel<<<(int)((t + 255) / 256), 256, 0, stream>>>(x,   y,   s1x, s1y, 721, 1440, 361, 720, 1);
        t = (long long)NIMG * 181 * 360;
        pool_kernel<<<(int)((t + 255) / 256), 256, 0, stream>>>(s1x, s1y, s2x, s2y, 361, 720, 181, 360, 0);
        t = (long long)NIMG * 91 * 180;
        pool_kernel<<<(int)((t + 255) / 256), 256, 0, stream>>>(s2x, s2y, s3x, s3y, 181, 360, 91, 180, 0);
        t = (long long)NIMG * 46 * 90;
        pool_kernel<<<(int)((t + 255) / 256), 256, 0, stream>>>(s3x, s3y, s4x, s4y, 91, 180, 46, 90, 0);
    }

    // SSIM at each scale (scale 0 normalizes raw inputs on the fly).
    ssim_kernel<<<dim3(23 * 23, NIMG), 256, 0, stream>>>(x,   y,   721, 1440, 711, 1430, 23, 1, acc + 0 * NIMG, acc + 1024 + 0 * NIMG);
    ssim_kernel<<<dim3(11 * 12, NIMG), 256, 0, stream>>>(s1x, s1y, 361,  720, 351,  710, 12, 0, acc + 1 * NIMG, acc + 1024 + 1 * NIMG);
    ssim_kernel<<<dim3( 6 *  6, NIMG), 256, 0, stream>>>(s2x, s2y, 181,  360, 171,  350,  6, 0, acc + 2 * NIMG, acc + 1024 + 2 * NIMG);
    ssim_kernel<<<dim3( 3 *  3, NIMG), 256, 0, stream>>>(s3x, s3y,  91,  180,  81,  170,  3, 0, acc + 3 * NIMG, acc + 1024 + 3 * NIMG);
    ssim_kernel<<<dim3( 2 *  2, NIMG), 256, 0, stream>>>(s4x, s4y,  46,   90,  36,   80,  2, 0, acc + 4 * NIMG, acc + 1024 + 4 * NIMG);

    final_kernel<<<1, 128, 0, stream>>>(acc, out);
}